// MultiHeadAttention_55353538511182
// MI455X (gfx1250) — compile-verified
//
#include <hip/hip_runtime.h>
#include <math.h>

// ---------------------------------------------------------------------------
// MultiHeadAttention forward for MI455X (gfx1250), fp32 via V_WMMA_F32_16X16X4_F32
//   B=2, S=2048, D=1024, H=16, Hd=64
//   out  = [B,S,D]    at d_out[0]
//   attn = [B,H,S,S]  at d_out[4194304]
// ---------------------------------------------------------------------------

#define BATCH    2
#define SEQ      2048
#define D_MODEL  1024
#define NUM_HEAD 16
#define HEAD_DIM 64
#define NTOK     (BATCH * SEQ)                                // 4096
#define PROJ     ((size_t)BATCH * NUM_HEAD * SEQ * HEAD_DIM)  // 4,194,304 floats

typedef float v2f __attribute__((ext_vector_type(2)));
typedef float v8f __attribute__((ext_vector_type(8)));

// LDS row stride for the 32 x 2048 score strip; +4 pad so row-strided column
// reads (2052 % 64 banks == 4) hit distinct banks.
#define SC_STRIDE  2052
#define SC_ROWS    32
#define RED_OFF    (SC_ROWS * SC_STRIDE)          // 65664
#define RMAX_OFF   (RED_OFF + 256)                // 65920
#define RSUM_OFF   (RMAX_OFF + 32)                // 65952
#define QLDS_OFF   (RSUM_OFF + 32)                // 65984
#define QLDS_STRIDE 68                            // 64 + 4 pad (bank spread)
#define SMEM_FLOATS (QLDS_OFF + SC_ROWS * QLDS_STRIDE)  // 68160 -> 272,640 B

static __device__ __forceinline__ v8f wmma4(v2f a, v2f b, v8f c) {
    return __builtin_amdgcn_wmma_f32_16x16x4_f32(
        /*neg_a=*/false, a, /*neg_b=*/false, b,
        /*c_mod=*/(short)0, c, /*reuse_a=*/false, /*reuse_b=*/false);
}

// ---------------------------------------------------------------------------
// Kernel 1: y = x @ W^T + b, scattered to [B,H,S,Hd].
// One wave computes one 32x32 output block (2x2 register tiling of 16x16
// WMMA tiles): 4 WMMAs per 4 b64 loads -> 2x L2-traffic reduction vs 16x16.
// ---------------------------------------------------------------------------
__global__ __launch_bounds__(32)
void qkv_proj_kernel(const float* __restrict__ x,
                     const float* __restrict__ w,
                     const float* __restrict__ bias,
                     float* __restrict__ dst) {
    const int lane = threadIdx.x & 31;
    const int l15  = lane & 15;
    const int half = lane >> 4;           // 0 or 1
    const int koff = half * 2;            // K sub-offset within a K=4 step

    const int tm = blockIdx.x;            // 32-row tile over N = 4096 tokens
    const int te = blockIdx.y;            // 32-col tile over E = 1024 features

    const float* __restrict__ xr0 = x + (size_t)(tm * 32 + l15) * D_MODEL + koff;
    const float* __restrict__ xr1 = xr0 + (size_t)16 * D_MODEL;
    const float* __restrict__ wr0 = w + (size_t)(te * 32 + l15) * D_MODEL + koff;
    const float* __restrict__ wr1 = wr0 + (size_t)16 * D_MODEL;

    v8f c00 = {}, c01 = {}, c10 = {}, c11 = {};
#pragma unroll 4
    for (int k = 0; k < D_MODEL; k += 4) {
        if ((k & 63) == 0) {              // gfx1250 global_prefetch_b8
            __builtin_prefetch(xr0 + k + 128, 0, 1);
            __builtin_prefetch(xr1 + k + 128, 0, 1);
            __builtin_prefetch(wr0 + k + 128, 0, 1);
            __builtin_prefetch(wr1 + k + 128, 0, 1);
        }
        v2f a0 = *(const v2f*)(xr0 + k);
        v2f a1 = *(const v2f*)(xr1 + k);
        v2f b0 = *(const v2f*)(wr0 + k);
        v2f b1 = *(const v2f*)(wr1 + k);
        c00 = wmma4(a0, b0, c00);
        c01 = wmma4(a0, b1, c01);
        c10 = wmma4(a1, b0, c10);
        c11 = wmma4(a1, b1, c11);
    }

    // C/D layout: lanes 0-15: VGPR i -> (M=i, N=lane); lanes 16-31: (M=i+8, N=lane-16)
    const int e0 = te * 32 + l15;
    const int e1 = e0 + 16;
    const float bv0 = bias[e0];
    const float bv1 = bias[e1];
#pragma unroll
    for (int i = 0; i < 8; ++i) {
        const int m    = i + half * 8;
        const int n0   = tm * 32 + m;          // token rows for c0x
        const int n1   = n0 + 16;              // token rows for c1x
        const int b0i  = n0 >> 11, s0 = n0 & (SEQ - 1);
        const int b1i  = n1 >> 11, s1 = n1 & (SEQ - 1);
        dst[(((size_t)b0i * NUM_HEAD + (e0 >> 6)) * SEQ + s0) * HEAD_DIM + (e0 & 63)] = c00[i] + bv0;
        dst[(((size_t)b0i * NUM_HEAD + (e1 >> 6)) * SEQ + s0) * HEAD_DIM + (e1 & 63)] = c01[i] + bv1;
        dst[(((size_t)b1i * NUM_HEAD + (e0 >> 6)) * SEQ + s1) * HEAD_DIM + (e0 & 63)] = c10[i] + bv0;
        dst[(((size_t)b1i * NUM_HEAD + (e1 >> 6)) * SEQ + s1) * HEAD_DIM + (e1 & 63)] = c11[i] + bv1;
    }
}

// ---------------------------------------------------------------------------
// Kernel 2: fused scores -> softmax -> attn store -> O = P@V.
// One workgroup (8 waves / 256 threads) handles 32 query rows of one (b,h).
// Full 32x2048 probability strip lives in LDS (~263KB of CDNA5's 320KB/WGP).
// Q block staged via gfx1250 async global->LDS DMA (ASYNCcnt path).
// ---------------------------------------------------------------------------
__global__ __launch_bounds__(256)
void attn_fused_kernel(const float* __restrict__ Q,
                       const float* __restrict__ K,
                       const float* __restrict__ V,
                       float* __restrict__ attn,
                       float* __restrict__ out) {
    extern __shared__ float smem[];

    const int tid  = threadIdx.x;
    const int lane = tid & 31;
    const int wv   = tid >> 5;                    // wave id 0..7
    const int l15  = lane & 15;
    const int half = lane >> 4;
    const int koff = half * 2;

    const int qt = blockIdx.x;                    // 32-row query tile, 0..63
    const int h  = blockIdx.y;
    const int b  = blockIdx.z;
    const size_t bh = (size_t)b * NUM_HEAD + h;

    const float* __restrict__ Qb = Q + bh * SEQ * HEAD_DIM;
    const float* __restrict__ Kb = K + bh * SEQ * HEAD_DIM;
    const float* __restrict__ Vb = V + bh * SEQ * HEAD_DIM;

    // ---- async-stage the 32x64 Q block into (padded) LDS --------------------
    // 32 rows x 16 float4-chunks = 512 chunks; 256 threads x 2 iterations.
#pragma unroll
    for (int it = 0; it < 2; ++it) {
        const int cidx = tid + it * 256;
        const int row  = cidx >> 4;               // 0..31
        const int col4 = cidx & 15;               // float4 index within row
        const unsigned lds_byte =
            (unsigned)((QLDS_OFF + row * QLDS_STRIDE + col4 * 4) * 4);
        const unsigned long long gaddr =
            (unsigned long long)(const void*)(Qb + (size_t)(qt * 32 + row) * HEAD_DIM + col4 * 4);
        asm volatile("global_load_async_to_lds_b128 %0, %1, off"
                     :: "v"(lds_byte), "v"(gaddr) : "memory");
    }
    asm volatile("s_wait_asynccnt 0x0" ::: "memory");
    __syncthreads();

    // ---- preload this wave's Q-tile A fragments from LDS --------------------
    const int qsub = wv >> 2;                     // 0/1: which 16-row half
    v2f aq[16];
    {
        const float* qp = smem + QLDS_OFF + (qsub * 16 + l15) * QLDS_STRIDE + koff;
#pragma unroll
        for (int kk = 0; kk < 16; ++kk) {
            aq[kk].x = qp[kk * 4];
            aq[kk].y = qp[kk * 4 + 1];
        }
    }

    // ---- Phase A: scores = (Q K^T) * rsqrt(Hd) into LDS ---------------------
    // 4 waves per qsub split the 128 key tiles.
    for (int kt = (wv & 3); kt < SEQ / 16; kt += 4) {
        const float* kp = Kb + (size_t)(kt * 16 + l15) * HEAD_DIM + koff;
        __builtin_prefetch(kp + 4 * 16 * HEAD_DIM, 0, 1);   // next key tile
        v8f c = {};
#pragma unroll
        for (int kk = 0; kk < 16; ++kk) {
            v2f bb = *(const v2f*)(kp + kk * 4);
            c = wmma4(aq[kk], bb, c);
        }
        const int col = kt * 16 + l15;
#pragma unroll
        for (int i = 0; i < 8; ++i)
            smem[(qsub * 16 + i + half * 8) * SC_STRIDE + col] = c[i] * 0.125f;
    }
    __syncthreads();

    // ---- Phase B: row-wise softmax over [32][2048] --------------------------
    float* red    = smem + RED_OFF;
    float* rowmax = smem + RMAX_OFF;
    float* rowsum = smem + RSUM_OFF;

    const int r = tid >> 3;                       // row 0..31 (8 threads/row)
    const int j = tid & 7;

    float m = -3.402823466e38f;
    for (int c0 = j; c0 < SEQ; c0 += 8) m = fmaxf(m, smem[r * SC_STRIDE + c0]);
    red[tid] = m;
    __syncthreads();
    if (tid < 32) {
        float mm = red[tid * 8];
#pragma unroll
        for (int t = 1; t < 8; ++t) mm = fmaxf(mm, red[tid * 8 + t]);
        rowmax[tid] = mm;
    }
    __syncthreads();

    const float rm = rowmax[r];
    float sacc = 0.f;
    for (int c0 = j; c0 < SEQ; c0 += 8) {
        float e = __expf(smem[r * SC_STRIDE + c0] - rm);
        smem[r * SC_STRIDE + c0] = e;
        sacc += e;
    }
    red[tid] = sacc;
    __syncthreads();
    if (tid < 32) {
        float ss = 0.f;
#pragma unroll
        for (int t = 0; t < 8; ++t) ss += red[tid * 8 + t];
        rowsum[tid] = ss;
    }
    __syncthreads();

    // normalize in LDS + coalesced store of attn (the dominant 536MB write)
    float* attn_base = attn + (bh * SEQ + (size_t)qt * 32) * SEQ;
    for (int idx = tid; idx < SC_ROWS * SEQ; idx += 256) {
        const int rr = idx >> 11;                 // / SEQ
        const int cc = idx & (SEQ - 1);
        const float p = smem[rr * SC_STRIDE + cc] * (1.0f / rowsum[rr]);
        smem[rr * SC_STRIDE + cc] = p;
        attn_base[(size_t)rr * SEQ + cc] = p;
    }
    __syncthreads();

    // ---- Phase C: O = P @ V --------------------------------------------------
    // 8 waves = 2 qsubs x 4 N-tiles; wave owns output cols [nt*16, nt*16+16).
    const int nt   = wv & 3;
    const int vcol = nt * 16 + l15;
    const int arow = qsub * 16 + l15;
    v8f co = {};
    for (int k = 0; k < SEQ; k += 4) {
        v2f a;
        a.x = smem[arow * SC_STRIDE + k + koff];      // pad -> bank-conflict free
        a.y = smem[arow * SC_STRIDE + k + koff + 1];
        const float* vp = Vb + (size_t)(k + koff) * HEAD_DIM + vcol;
        v2f bb;
        bb.x = vp[0];
        bb.y = vp[HEAD_DIM];
        co = wmma4(a, bb, co);
    }
#pragma unroll
    for (int i = 0; i < 8; ++i) {
        const int s = qt * 32 + qsub * 16 + i + half * 8;
        const int e = h * HEAD_DIM + nt * 16 + l15;
        out[((size_t)b * SEQ + s) * D_MODEL + e] = co[i];
    }
}

// ---------------------------------------------------------------------------
extern "C" void kernel_launch(void* const* d_in, const int* in_sizes, int n_in,
                              void* d_out, int out_size, void* d_ws, size_t ws_size,
                              hipStream_t stream) {
    const float* x  = (const float*)d_in[0];
    const float* wq = (const float*)d_in[1];
    const float* bq = (const float*)d_in[2];
    const float* wk = (const float*)d_in[3];
    const float* bk = (const float*)d_in[4];
    const float* wv = (const float*)d_in[5];
    const float* bv = (const float*)d_in[6];

    float* out  = (float*)d_out;
    float* attn = out + (size_t)BATCH * SEQ * D_MODEL;

    float* qws = (float*)d_ws;          // [B,H,S,Hd] each, 16MB apiece
    float* kws = qws + PROJ;
    float* vws = kws + PROJ;

    dim3 g1(NTOK / 32, D_MODEL / 32);   // 128 x 32 blocks, 1 wave each
    qkv_proj_kernel<<<g1, 32, 0, stream>>>(x, wq, bq, qws);
    qkv_proj_kernel<<<g1, 32, 0, stream>>>(x, wk, bk, kws);
    qkv_proj_kernel<<<g1, 32, 0, stream>>>(x, wv, bv, vws);

    dim3 g2(SEQ / 32, NUM_HEAD, BATCH); // 64 x 16 x 2 workgroups
    const size_t smem_bytes = SMEM_FLOATS * sizeof(float);  // ~272.6 KB LDS
    attn_fused_kernel<<<g2, 256, smem_bytes, stream>>>(qws, kws, vws, attn, out);
}